// GraphAttentionLayer_64939905515997
// MI455X (gfx1250) — compile-verified
//
#include <hip/hip_runtime.h>
#include <hip/hip_bf16.h>
#include <math.h>

// Problem constants (from reference)
#define BATCH 4
#define SEQ   336   // 21 * 16
#define INF   512   // K for projection GEMMs
#define NOUT  256   // out features / graph nodes, 16 * 16

typedef __attribute__((ext_vector_type(2))) float v2f;
typedef __attribute__((ext_vector_type(8))) float v8f;

// ---------------------------------------------------------------------------
// Kernel 1: g_r[b,n,s] = sum_k W_r[n,k] * x[b,s,k]     (stored [b][n][s])
//           g_l[b,n,s] = sum_k W_l[n,k] * x[b,s,k]     (stored transposed [b][s][n])
// One 16x16 output tile per wave via V_WMMA_F32_16X16X4_F32 (exact fp32 FMA).
// Fragment layout per CDNA5 ISA 7.12.2 (32-bit A 16x4):
//   lane<16 : M = lane,     holds K = {k, k+1}
//   lane>=16: M = lane-16,  holds K = {k+2, k+3}
// B (4x16) mirrored with lane -> N, VGPR -> K.  Each lane loads contiguous float2.
// ---------------------------------------------------------------------------
__global__ __launch_bounds__(128)
void gat_proj_wmma(const float* __restrict__ x,
                   const float* __restrict__ Wl,
                   const float* __restrict__ Wr,
                   float* __restrict__ g_r,
                   float* __restrict__ g_lT) {
  const int lane = threadIdx.x & 31;
  const int wave = threadIdx.x >> 5;
  const int l16  = lane & 15;
  const int hi   = lane >> 4;

  const int m0  = (blockIdx.x * 4 + wave) * 16;   // output-feature tile
  const int s0  = blockIdx.y * 16;                // sequence tile
  const int b   = blockIdx.z & 3;
  const int sel = blockIdx.z >> 2;                // 0 -> W_r/g_r, 1 -> W_l/g_lT

  const float* W = sel ? Wl : Wr;
  // A fragment base: W[m0+l16][2*hi + k..k+1]
  const float* Arow = W + (size_t)(m0 + l16) * INF + 2 * hi;
  // B fragment base: x[b][s0+l16][2*hi + k..k+1]
  const float* Brow = x + ((size_t)b * SEQ + (s0 + l16)) * INF + 2 * hi;

  v8f c = {0.f, 0.f, 0.f, 0.f, 0.f, 0.f, 0.f, 0.f};

#pragma unroll 8
  for (int k = 0; k < INF; k += 4) {
    v2f af = *(const v2f*)(Arow + k);
    v2f bf = *(const v2f*)(Brow + k);
    c = __builtin_amdgcn_wmma_f32_16x16x4_f32(false, af, false, bf,
                                              (short)0, c, false, false);
  }

  // C/D layout: VGPR r, lane -> element (M = r + 8*hi, N = l16)
  if (sel == 0) {
    float* outp = g_r + (size_t)b * NOUT * SEQ;
#pragma unroll
    for (int r = 0; r < 8; ++r) {
      const int m = m0 + r + 8 * hi;
      outp[(size_t)m * SEQ + (s0 + l16)] = c[r];   // coalesced in s
    }
  } else {
    float* outp = g_lT + (size_t)b * SEQ * NOUT;
#pragma unroll
    for (int r = 0; r < 8; ++r) {
      const int m = m0 + r + 8 * hi;
      outp[(size_t)(s0 + l16) * NOUT + m] = c[r];  // transposed store
    }
  }
}

// ---------------------------------------------------------------------------
// Kernel 2: per (b,i) row — e[j] = sum_s leaky_relu(g_r[i,s]+g_l[j,s], 0.2)*a[s],
// adjacency mask, block-wide softmax over j.  256 threads (8 waves), thread = j.
// g_lT reads are lane-coalesced (consecutive j within a cacheline).
// ---------------------------------------------------------------------------
__global__ __launch_bounds__(NOUT)
void gat_attn_softmax(const float* __restrict__ g_r,
                      const float* __restrict__ g_lT,
                      const int* __restrict__ adj,
                      const float* __restrict__ avec,
                      float* __restrict__ att) {
  __shared__ float gr_row[SEQ];
  __shared__ float a_sh[SEQ];
  __shared__ float red[NOUT];

  const int j = threadIdx.x;
  const int i = blockIdx.x;
  const int b = blockIdx.y;

  const float* grp = g_r + ((size_t)b * NOUT + i) * SEQ;
  for (int s = j; s < SEQ; s += NOUT) {
    gr_row[s] = grp[s];
    a_sh[s]   = avec[s];
  }
  __syncthreads();

  const float* gl = g_lT + (size_t)b * SEQ * NOUT + j;
  float acc = 0.f;
#pragma unroll 4
  for (int s = 0; s < SEQ; ++s) {
    const float v  = gr_row[s] + gl[(size_t)s * NOUT];
    const float lr = fmaxf(v, 0.2f * v);           // leaky_relu(v, 0.2)
    acc = fmaf(lr, a_sh[s], acc);
  }

  const float e = (adj[i * NOUT + j] > 0) ? acc : -9e15f;

  // block max
  red[j] = e;
  __syncthreads();
#pragma unroll
  for (int off = NOUT / 2; off >= 1; off >>= 1) {
    if (j < off) red[j] = fmaxf(red[j], red[j + off]);
    __syncthreads();
  }
  const float mx = red[0];
  __syncthreads();

  const float ex = __expf(e - mx);
  red[j] = ex;
  __syncthreads();
#pragma unroll
  for (int off = NOUT / 2; off >= 1; off >>= 1) {
    if (j < off) red[j] += red[j + off];
    __syncthreads();
  }
  const float inv = 1.0f / red[0];

  att[((size_t)b * NOUT + i) * NOUT + j] = ex * inv;
}

// ---------------------------------------------------------------------------
// Kernel 3: h[b,i,s] = sum_j att[b,i,j] * g_r[b,j,s]   (M=i, K=j, N=s)
// Same f32 WMMA tiling as kernel 1.
// ---------------------------------------------------------------------------
__global__ __launch_bounds__(128)
void gat_out_wmma(const float* __restrict__ att,
                  const float* __restrict__ g_r,
                  float* __restrict__ out) {
  const int lane = threadIdx.x & 31;
  const int wave = threadIdx.x >> 5;
  const int l16  = lane & 15;
  const int hi   = lane >> 4;

  const int m0 = (blockIdx.x * 4 + wave) * 16;  // i tile
  const int s0 = blockIdx.y * 16;               // s tile
  const int b  = blockIdx.z;

  const float* Ap = att + ((size_t)b * NOUT + (m0 + l16)) * NOUT + 2 * hi;
  const float* Bp = g_r + (size_t)b * NOUT * SEQ + (s0 + l16);

  v8f c = {0.f, 0.f, 0.f, 0.f, 0.f, 0.f, 0.f, 0.f};

#pragma unroll 8
  for (int k = 0; k < NOUT; k += 4) {
    v2f af = *(const v2f*)(Ap + k);             // att[i, k+2*hi .. +1], contiguous
    const int kk = k + 2 * hi;
    v2f bf;
    bf.x = Bp[(size_t)kk * SEQ];                // g_r[kk,   s0+l16] (lane-coalesced)
    bf.y = Bp[(size_t)(kk + 1) * SEQ];          // g_r[kk+1, s0+l16]
    c = __builtin_amdgcn_wmma_f32_16x16x4_f32(false, af, false, bf,
                                              (short)0, c, false, false);
  }

#pragma unroll
  for (int r = 0; r < 8; ++r) {
    const int m = m0 + r + 8 * hi;
    out[((size_t)b * NOUT + m) * SEQ + (s0 + l16)] = c[r];
  }
}

// ---------------------------------------------------------------------------
// Launch
// inputs: 0=x [4,336,512] f32, 1=adj [256,256] i32, 2=W_l [256,512] f32,
//         3=W_r [256,512] f32, 4=a [336] f32
// output: h_prime [4,256,336] f32
// ---------------------------------------------------------------------------
extern "C" void kernel_launch(void* const* d_in, const int* in_sizes, int n_in,
                              void* d_out, int out_size, void* d_ws, size_t ws_size,
                              hipStream_t stream) {
  const float* x    = (const float*)d_in[0];
  const int*   adj  = (const int*)d_in[1];
  const float* W_l  = (const float*)d_in[2];
  const float* W_r  = (const float*)d_in[3];
  const float* avec = (const float*)d_in[4];
  float* out = (float*)d_out;

  float* g_r  = (float*)d_ws;                               // [B][NOUT][SEQ]
  float* g_lT = g_r + (size_t)BATCH * NOUT * SEQ;           // [B][SEQ][NOUT]
  float* att  = g_lT + (size_t)BATCH * SEQ * NOUT;          // [B][NOUT][NOUT]

  // K1: 16 m-tiles (4 blocks x 4 waves), 21 s-tiles, 4 batches x 2 weights
  dim3 g1(4, SEQ / 16, BATCH * 2);
  gat_proj_wmma<<<g1, 128, 0, stream>>>(x, W_l, W_r, g_r, g_lT);

  // K2: one block per (b, i) attention row
  dim3 g2(NOUT, BATCH, 1);
  gat_attn_softmax<<<g2, NOUT, 0, stream>>>(g_r, g_lT, adj, avec, att);

  // K3: output GEMM
  dim3 g3(4, SEQ / 16, BATCH);
  gat_out_wmma<<<g3, 128, 0, stream>>>(att, g_r, out);
}